// HashGridEmbedder_20607253086643
// MI455X (gfx1250) — compile-verified
//
#include <hip/hip_runtime.h>

// ---------------------------------------------------------------------------
// Instant-NGP hash-grid forward for MI455X (gfx1250, wave32).
//
// Perf model: 42MB embedding table resident in 192MB L2; kernel is bound by
// L2 gather request rate / latency (~128 scattered 8B gathers/point), not
// FLOPs, so no WMMA (a (1x8)x(8x2) blend wastes >97% of a 16x16xK WMMA and
// costs layout shuffles). Levers used instead:
//  * level-0 table (4913 entries, 39.3KB) cached in LDS via CDNA5
//    global_load_async_to_lds_b64 (ASYNCcnt), gathered with ds ops
//  * 256-thread blocks: 39.3KB table shared by 8 waves; 76.2KB/block ->
//    4 blocks/WGP = 32 waves/WGP = 8 waves/SIMD of gather latency hiding
//  * xyz in / out streams non-temporal so they never evict the table from L2
//  * output staged per-wave in LDS, then written fully coalesced (NT)
//  * levels 7..15: table size 2^19 -> pure u32 hash + mask (XOR is bitwise)
//  * levels 0..6 : h<2^38 -> split-word modulo: all-32-bit magic urems
// ---------------------------------------------------------------------------

namespace {

constexpr int      kLevels = 16;
constexpr int      kOutF   = 35;   // 3 (xyz) + 16 levels * 2 feats
constexpr int      kPadF   = 36;   // padded LDS row stride (floats)
constexpr int      kBlock  = 256;  // 8 waves of 32
constexpr int      kL0     = 4913; // level-0 entries (17^3)
constexpr unsigned kPrimeY = 2654435761u;
constexpr unsigned kPrimeZ = 805459861u;

constexpr int kRes[kLevels] = {16, 20, 25, 32, 40, 50, 64, 80,
                               101, 128, 161, 203, 256, 322, 406, 512};
constexpr unsigned kSize[kLevels] = {
    4913u, 9261u, 17576u, 35937u, 68921u, 132651u, 274625u,
    524288u, 524288u, 524288u, 524288u, 524288u, 524288u, 524288u, 524288u, 524288u};
constexpr unsigned kOff[kLevels] = {
    0u, 4913u, 14174u, 31750u, 67687u, 136608u, 269259u, 543884u,
    1068172u, 1592460u, 2116748u, 2641036u, 3165324u, 3689612u, 4213900u, 4738188u};

} // namespace

__global__ __launch_bounds__(kBlock)
void hashgrid_fwd_kernel(const float* __restrict__ xyz,
                         const float* __restrict__ emb,
                         float* __restrict__ out,
                         int nPoints)
{
    typedef unsigned long long u64;

    // 39,304B level-0 cache + 36,864B output stage = 76.2KB (WGP LDS is 320KB)
    __shared__ __align__(16) float s_l0[2 * kL0];
    __shared__ __align__(16) float s_stage[kBlock * kPadF];

    // ---- fill the level-0 LDS cache (coalesced 8B/lane stream) ----
#if __has_builtin(__builtin_amdgcn_global_load_async_to_lds_b64)
    {
        // CDNA5 async copy: memory -> LDS without a VGPR round trip (ASYNCcnt).
        typedef int v2i __attribute__((__vector_size__(2 * sizeof(int))));
        typedef __attribute__((address_space(1))) v2i gv2i;  // global
        typedef __attribute__((address_space(3))) v2i lv2i;  // LDS
        gv2i* g = (gv2i*)emb;      // level 0 starts at offset 0
        lv2i* l = (lv2i*)s_l0;
        for (int i = (int)threadIdx.x; i < kL0; i += kBlock)
            __builtin_amdgcn_global_load_async_to_lds_b64(g + i, l + i, 0, 0);
        asm volatile("s_wait_asynccnt 0" ::: "memory");
    }
#else
    {
        const float2* src = reinterpret_cast<const float2*>(emb);  // level 0 at offset 0
        float2*       dst = reinterpret_cast<float2*>(s_l0);
        for (int i = (int)threadIdx.x; i < kL0; i += kBlock) dst[i] = src[i];
    }
#endif
    __syncthreads();

    const int lane = (int)(threadIdx.x & 31u);
    const int wave = (int)(threadIdx.x >> 5u);
    const int p0   = (blockIdx.x * (kBlock / 32) + wave) * 32;  // 32-pt tile base
    if (p0 >= nPoints) return;                                  // whole-wave guard

    const int p  = p0 + lane;
    const int pc = (p < nPoints) ? p : (nPoints - 1);           // clamp (B%32==0 anyway)

    // streaming input: NT so it can't evict the embedding working set
    const float x = __builtin_nontemporal_load(xyz + 3 * (size_t)pc + 0);
    const float y = __builtin_nontemporal_load(xyz + 3 * (size_t)pc + 1);
    const float z = __builtin_nontemporal_load(xyz + 3 * (size_t)pc + 2);

    // norm = (xyz - (-0.75)) / 1.5, reference op order in f32
    const float n0 = (x + 0.75f) / 1.5f;
    const float n1 = (y + 0.75f) / 1.5f;
    const float n2 = (z + 0.75f) / 1.5f;

    float* myRow = s_stage + (int)threadIdx.x * kPadF;
    myRow[0] = x; myRow[1] = y; myRow[2] = z;

#pragma unroll
    for (int lvl = 0; lvl < kLevels; ++lvl) {
        const float r   = (float)kRes[lvl];
        const int   rm1 = kRes[lvl] - 1;

        const float px = n0 * r, py = n1 * r, pz = n2 * r;
        int bx = (int)floorf(px); bx = bx < 0 ? 0 : (bx > rm1 ? rm1 : bx);
        int by = (int)floorf(py); by = by < 0 ? 0 : (by > rm1 ? rm1 : by);
        int bz = (int)floorf(pz); bz = bz < 0 ? 0 : (bz > rm1 ? rm1 : bz);

        const float fx = px - (float)bx, fy = py - (float)by, fz = pz - (float)bz;
        const float gx = 1.0f - fx,      gy = 1.0f - fy,      gz = 1.0f - fz;
        // corner index c: bit2 = x-corner, bit1 = y-corner, bit0 = z-corner
        const float wxy[4] = {gx * gy, gx * fy, fx * gy, fx * fy};

        unsigned idx[8];
        if (kSize[lvl] == 524288u) {
            // power-of-two table: XOR is bitwise, mask needs only low 32 bits
            const unsigned hx0 = (unsigned)bx,           hx1 = hx0 + 1u;
            const unsigned hy0 = (unsigned)by * kPrimeY, hy1 = hy0 + kPrimeY;
            const unsigned hz0 = (unsigned)bz * kPrimeZ, hz1 = hz0 + kPrimeZ;
            const unsigned hyz[4] = {hy0 ^ hz0, hy0 ^ hz1, hy1 ^ hz0, hy1 ^ hz1};
#pragma unroll
            for (int c = 0; c < 8; ++c) {
                const unsigned h = ((c & 4) ? hx1 : hx0) ^ hyz[c & 3];
                idx[c] = (h & 524287u) + kOff[lvl];
            }
        } else {
            // dense level: h = (hi<<32)|lo with hi = hi(y*P1)^hi(z*P2) < 2^7.
            // h mod m = ((hi*2^32 mod m) + lo mod m) mod m  -- all 32-bit ops,
            // constant divisors after full unroll -> magic-number urems.
            const unsigned m   = kSize[lvl];
            const unsigned R32 = (unsigned)(4294967296ull % (u64)m);  // folds
            const u64 hy0 = (u64)(unsigned)by * kPrimeY, hy1 = hy0 + kPrimeY;
            const u64 hz0 = (u64)(unsigned)bz * kPrimeZ, hz1 = hz0 + kPrimeZ;
            const unsigned hiy0 = (unsigned)(hy0 >> 32), hiy1 = (unsigned)(hy1 >> 32);
            const unsigned hiz0 = (unsigned)(hz0 >> 32), hiz1 = (unsigned)(hz1 >> 32);
            // 4 y/z high-word combos, premultiplied & premodded
            const unsigned hm[4] = {((hiy0 ^ hiz0) * R32) % m,
                                    ((hiy0 ^ hiz1) * R32) % m,
                                    ((hiy1 ^ hiz0) * R32) % m,
                                    ((hiy1 ^ hiz1) * R32) % m};
            const unsigned ly0 = (unsigned)hy0, ly1 = (unsigned)hy1;
            const unsigned lz0 = (unsigned)hz0, lz1 = (unsigned)hz1;
            const unsigned lyz[4] = {ly0 ^ lz0, ly0 ^ lz1, ly1 ^ lz0, ly1 ^ lz1};
            const unsigned lx0 = (unsigned)bx, lx1 = lx0 + 1u;
#pragma unroll
            for (int c = 0; c < 8; ++c) {
                const unsigned lo = ((c & 4) ? lx1 : lx0) ^ lyz[c & 3];
                const unsigned s  = hm[c & 3] + lo % m;          // < 2m
                idx[c] = (s >= m ? s - m : s) + kOff[lvl];
            }
        }

        float a0 = 0.0f, a1 = 0.0f;
        if (lvl == 0) {
            // gather from LDS cache (ds_load_b64), off the VMEM critical path
            const float2* tab = reinterpret_cast<const float2*>(s_l0);
#pragma unroll
            for (int c = 0; c < 8; ++c) {
                const float2 e = tab[idx[c]];
                const float  w = wxy[c >> 1] * ((c & 1) ? fz : gz);
                a0 = fmaf(w, e.x, a0);
                a1 = fmaf(w, e.y, a1);
            }
        } else {
            // gather from the L2-resident table (default RT temporal policy)
#pragma unroll
            for (int c = 0; c < 8; ++c) {
                const float2 e = *reinterpret_cast<const float2*>(emb + 2u * (size_t)idx[c]);
                const float  w = wxy[c >> 1] * ((c & 1) ? fz : gz);
                a0 = fmaf(w, e.x, a0);
                a1 = fmaf(w, e.y, a1);
            }
        }
        myRow[3 + 2 * lvl]     = a0;
        myRow[3 + 2 * lvl + 1] = a1;
    }

    // Intra-wave LDS handoff: LDS is in-order per wave in HW; the asm is a
    // compiler reorder fence + explicit CDNA5 split-counter drain.
    asm volatile("s_wait_dscnt 0" ::: "memory");

    // ---- coalesced NT write-back: this wave's 32 points = 1120 contiguous floats
    float*       dst = out + (size_t)p0 * kOutF;
    const float* src = s_stage + wave * 32 * kPadF;
    const int nvalid = nPoints - p0;
    const int lim    = (nvalid >= 32 ? 32 : nvalid) * kOutF;
#pragma unroll
    for (int k = 0; k < kOutF; ++k) {
        const int linear = k * 32 + lane;          // 0..1119
        if (linear < lim) {
            const int pt = linear / kOutF;         // constant-divisor magic
            const int cp = linear - pt * kOutF;
            __builtin_nontemporal_store(src[pt * kPadF + cp], dst + linear);
        }
    }
}

extern "C" void kernel_launch(void* const* d_in, const int* in_sizes, int n_in,
                              void* d_out, int out_size, void* d_ws, size_t ws_size,
                              hipStream_t stream)
{
    const float* xyz = (const float*)d_in[0];   // (B,3) f32
    const float* emb = (const float*)d_in[1];   // (5262476,2) f32
    float*       out = (float*)d_out;           // (B,35) f32

    const int nPoints = in_sizes[0] / 3;
    const int blocks  = (nPoints + kBlock - 1) / kBlock;
    hashgrid_fwd_kernel<<<blocks, kBlock, 0, stream>>>(xyz, emb, out, nPoints);
}